// Attention_72696616452632
// MI455X (gfx1250) — compile-verified
//
#include <hip/hip_runtime.h>

typedef float v2f __attribute__((ext_vector_type(2)));
typedef float v4f __attribute__((ext_vector_type(4)));
typedef float v8f __attribute__((ext_vector_type(8)));

// Problem constants (from reference): B=16, T_his=4095, T=4096, hid=2048, H=16, dh=128
constexpr int    kBatch = 16;
constexpr int    kThis  = 4095;
constexpr int    kTtot  = 4096;
constexpr int    kHid   = 2048;
constexpr int    kNH    = 16;
constexpr int    kDH    = 128;
constexpr size_t kPerBatchOut = (size_t)kTtot * kHid;          // 8388608
constexpr size_t kTensorOut   = (size_t)kBatch * kPerBatchOut; // 134217728

// ---------------------------------------------------------------------------
// Skinny GEMM: C[16, Nc] = A[16, K] @ W[:, wcol0 + 0..Nc) + bias[wcol0 + ...]
// One wave32 per 16-wide output tile using V_WMMA_F32_16X16X4_F32.
//   A (16x4 f32): lanes 0-15 hold M=lane K={0,1}; lanes 16-31 hold K={2,3}
//   B (4x16 f32): lanes 0-15 hold N=lane K={0,1}; lanes 16-31 hold K={2,3}
//   C (16x16 f32): lanes 0-15 N=lane M=0..7 in v[0..7]; lanes 16-31 M=8..15
// ---------------------------------------------------------------------------
__global__ __launch_bounds__(32) void wmma_gemm16(
    const float* __restrict__ A, const float* __restrict__ W,
    const float* __restrict__ bias, float* __restrict__ C,
    int K, int Nw, int Nc, int wcol0)
{
    const int lane = threadIdx.x & 31;
    const int m    = lane & 15;            // row (A) / col (B) index within tile
    const int kh   = (lane >> 4) << 1;     // 0 or 2: which K-pair this half-wave holds
    const int n0   = blockIdx.x * 16;

    const float* arow = A + (size_t)m * K + kh;
    const float* bcol = W + (size_t)kh * Nw + (size_t)(wcol0 + n0 + m);

    v8f acc = {};
    for (int k0 = 0; k0 < K; k0 += 4) {
        v2f a, b;
        a.x = arow[k0];
        a.y = arow[k0 + 1];
        b.x = bcol[(size_t)k0 * Nw];
        b.y = bcol[(size_t)(k0 + 1) * Nw];
        acc = __builtin_amdgcn_wmma_f32_16x16x4_f32(
            /*neg_a=*/false, a, /*neg_b=*/false, b,
            /*c_mod=*/(short)0, acc, /*reuse_a=*/false, /*reuse_b=*/false);
    }

    const int   col   = n0 + m;
    const float bv    = bias[wcol0 + col];
    const int   rbase = (lane >> 4) * 8;
#pragma unroll
    for (int r = 0; r < 8; ++r)
        C[(size_t)(rbase + r) * Nc + col] = acc[r] + bv;
}

// ---------------------------------------------------------------------------
// K_out[b, h, t, :] = (t < 4095) ? K_his[b, t, h*128 : (h+1)*128]
//                                : k_new[b, h*128 : (h+1)*128]
// One wave moves one dh=128-float chunk (32 lanes x float4 = 512B, coalesced
// on both ends). Non-temporal: streamed once, don't pollute L2.
// ---------------------------------------------------------------------------
__global__ __launch_bounds__(256) void copy_k(
    const float* __restrict__ Khis, const float* __restrict__ kvnew,
    float* __restrict__ Kout)
{
    const int  lane   = threadIdx.x & 31;
    const int  wave   = threadIdx.x >> 5;
    const long total  = (long)kBatch * kNH * kTtot;            // 1048576 chunks
    const long nwaves = (long)gridDim.x * (blockDim.x >> 5);

    for (long c = (long)blockIdx.x * (blockDim.x >> 5) + wave; c < total; c += nwaves) {
        const int t = (int)(c & (kTtot - 1));
        const int h = (int)((c >> 12) & (kNH - 1));
        const int b = (int)(c >> 16);

        const v4f* src;
        if (t < kThis)
            src = (const v4f*)(Khis + (size_t)b * kThis * kHid + (size_t)t * kHid + h * kDH) + lane;
        else
            src = (const v4f*)(kvnew + (size_t)b * (2 * kHid) + h * kDH) + lane; // k half

        v4f val = __builtin_nontemporal_load(src);

        v4f* dst = (v4f*)(Kout + (size_t)b * kPerBatchOut
                               + ((size_t)h * kTtot + t) * kDH) + lane;
        __builtin_nontemporal_store(val, dst);
    }
}

// ---------------------------------------------------------------------------
// Same permuted copy for V, with the column-sum (v_next precursor) fused in:
// block owns (b, h, 512-t slab); each of 8 waves accumulates 64 t's in float4
// registers, LDS ds_add_f32 combines waves, 128 global atomics per block.
// ---------------------------------------------------------------------------
__global__ __launch_bounds__(256) void copy_v_reduce(
    const float* __restrict__ Vhis, const float* __restrict__ kvnew,
    float* __restrict__ Vout, float* __restrict__ vsum)
{
    __shared__ float s[kDH];
    const int lane = threadIdx.x & 31;
    const int wave = threadIdx.x >> 5;
    const int bh   = blockIdx.x >> 3;       // 0..255
    const int tb   = blockIdx.x & 7;        // 8 slabs of 512 t's
    const int b    = bh >> 4;
    const int h    = bh & 15;

    if (threadIdx.x < kDH) s[threadIdx.x] = 0.0f;
    __syncthreads();

    v4f acc = {0.f, 0.f, 0.f, 0.f};
    const int t0 = tb * 512 + wave * 64;
#pragma unroll 4
    for (int i = 0; i < 64; ++i) {
        const int t = t0 + i;
        const v4f* src;
        if (t < kThis)
            src = (const v4f*)(Vhis + (size_t)b * kThis * kHid + (size_t)t * kHid + h * kDH) + lane;
        else
            src = (const v4f*)(kvnew + (size_t)b * (2 * kHid) + kHid + h * kDH) + lane; // v half

        v4f val = __builtin_nontemporal_load(src);
        acc.x += val.x; acc.y += val.y; acc.z += val.z; acc.w += val.w;

        v4f* dst = (v4f*)(Vout + (size_t)b * kPerBatchOut
                               + ((size_t)h * kTtot + t) * kDH) + lane;
        __builtin_nontemporal_store(val, dst);
    }

    atomicAdd(&s[lane * 4 + 0], acc.x);     // ds_add_f32
    atomicAdd(&s[lane * 4 + 1], acc.y);
    atomicAdd(&s[lane * 4 + 2], acc.z);
    atomicAdd(&s[lane * 4 + 3], acc.w);
    __syncthreads();

    if (threadIdx.x < kDH)
        atomicAdd(&vsum[(size_t)b * kHid + h * kDH + threadIdx.x], s[threadIdx.x]);
}

// ---------------------------------------------------------------------------
extern "C" void kernel_launch(void* const* d_in, const int* in_sizes, int n_in,
                              void* d_out, int out_size, void* d_ws, size_t ws_size,
                              hipStream_t stream)
{
    const float* x     = (const float*)d_in[0];
    const float* Khis  = (const float*)d_in[1];
    const float* Vhis  = (const float*)d_in[2];
    const float* Wattn = (const float*)d_in[3];
    const float* battn = (const float*)d_in[4];
    const float* Wproj = (const float*)d_in[5];
    const float* bproj = (const float*)d_in[6];
    // d_in[7]: n_head == 16 (hardcoded)

    float* out   = (float*)d_out;
    float* Kout  = out;
    float* Vout  = out + kTensorOut;
    float* vnext = out + 2 * kTensorOut;

    float* kvnew = (float*)d_ws;                       // [16, 4096]: k | v
    float* vsum  = kvnew + (size_t)kBatch * 2 * kHid;  // [16, 2048]

    // 1) [k|v] = x @ W_attn[:, 2048:6144] + b_attn[2048:6144]  (q is dead: softmax(size-1)=1)
    wmma_gemm16<<<(2 * kHid) / 16, 32, 0, stream>>>(
        x, Wattn, battn, kvnew, kHid, 3 * kHid, 2 * kHid, kHid);

    // 2) zero the reduction buffer (ws is poisoned by harness)
    hipMemsetAsync(vsum, 0, (size_t)kBatch * kHid * sizeof(float), stream);

    // 3) K cache: append + [T,H,dh] -> [H,T,dh] permuted streaming copy
    copy_k<<<2048, 256, 0, stream>>>(Khis, kvnew, Kout);

    // 4) V cache: same copy, fused with sum over T (att is all-ones)
    copy_v_reduce<<<kBatch * kNH * 8, 256, 0, stream>>>(Vhis, kvnew, Vout, vsum);

    // 5) v_next = vsum @ W_proj + b_proj
    wmma_gemm16<<<kHid / 16, 32, 0, stream>>>(
        vsum, Wproj, bproj, vnext, kHid, kHid, kHid, 0);
}